// IHPool_15178414424100
// MI455X (gfx1250) — compile-verified
//
#include <hip/hip_runtime.h>
#include <hip/hip_bf16.h>
#include <math.h>

// Problem constants (from reference)
#define N_PTS   65536
#define C_DIM   256
#define STRIDE_ 20
#define K_CENT  3277   // ceil(65536/20)

typedef __attribute__((ext_vector_type(2))) float v2f;
typedef __attribute__((ext_vector_type(8))) float v8f;

// ---------------------------------------------------------------------------
// CDNA5 async global->LDS copy helpers (ASYNCcnt path). The builtins take
// typed address-space pointers: b32 -> (AS1 int*, AS3 int*, imm, imm),
// b128 -> (AS1 v4i*, AS3 v4i*, imm, imm) where v4i = vector_size(16) int
// (exact types confirmed by the compiler diagnostic in the previous round).
// Guarded so the host pass / builtin-less toolchains fall back to a plain
// synchronous copy.
// ---------------------------------------------------------------------------
#if defined(__HIP_DEVICE_COMPILE__) && \
    __has_builtin(__builtin_amdgcn_global_load_async_to_lds_b32) && \
    __has_builtin(__builtin_amdgcn_global_load_async_to_lds_b128)
#define HAVE_ASYNC_LDS 1
typedef int async_v4i __attribute__((vector_size(16)));
typedef __attribute__((address_space(1))) int*       gas32p;
typedef __attribute__((address_space(3))) int*       las32p;
typedef __attribute__((address_space(1))) async_v4i* gas128p;
typedef __attribute__((address_space(3))) async_v4i* las128p;
__device__ __forceinline__ void async_g2l_b32(const void* g, void* l) {
    __builtin_amdgcn_global_load_async_to_lds_b32((gas32p)g, (las32p)l, 0, 0);
}
__device__ __forceinline__ void async_g2l_b128(const void* g, void* l) {
    __builtin_amdgcn_global_load_async_to_lds_b128((gas128p)g, (las128p)l, 0, 0);
}
#else
#define HAVE_ASYNC_LDS 0
__device__ __forceinline__ void async_g2l_b32(const void* g, void* l) {
    *(unsigned int*)l = *(const unsigned int*)g;
}
__device__ __forceinline__ void async_g2l_b128(const void* g, void* l) {
    *(uint4*)l = *(const uint4*)g;
}
#endif

__device__ __forceinline__ void wait_async0() {
#if defined(__HIP_DEVICE_COMPILE__)
#if __has_builtin(__builtin_amdgcn_s_wait_asynccnt)
    __builtin_amdgcn_s_wait_asynccnt(0);
#elif HAVE_ASYNC_LDS
    asm volatile("s_wait_asynccnt 0x0" ::: "memory");
#endif
#endif
}

// ---------------------------------------------------------------------------
// Kernel 0: zero a 32-bit region (rank array / accumulators)
// ---------------------------------------------------------------------------
__global__ void k_zero(unsigned int* __restrict__ p, int n) {
    int i = blockIdx.x * blockDim.x + threadIdx.x;
    if (i < n) p[i] = 0u;
}

// ---------------------------------------------------------------------------
// Kernel 1: fitness = tanh((x . w1) / ||w1||) via V_WMMA_F32_16X16X4_F32.
// Each wave computes 16 rows. A = x[16 rows][k..k+3] (f32 16x4 layout:
// VGPR0 = K={0|2}, VGPR1 = K={1|3} split by lane half). B = w1[k..k+3]
// broadcast across all 16 columns (every column of D equals the dot product).
// Also emits the order-preserving sort key for the rank pass.
// ---------------------------------------------------------------------------
__global__ void k_fitness(const float* __restrict__ x,
                          const float* __restrict__ w1,
                          float* __restrict__ fitness,
                          unsigned int* __restrict__ keys) {
    const int lane = threadIdx.x & 31;
    const int wave = threadIdx.x >> 5;
    const int rowBase = (blockIdx.x * (blockDim.x >> 5) + wave) * 16;

    // ||w1||^(-1): lane-parallel partial sums + wave32 xor-shuffle reduce
    float s = 0.f;
    for (int j = lane; j < C_DIM; j += 32) { float w = w1[j]; s += w * w; }
    for (int off = 16; off > 0; off >>= 1) s += __shfl_xor(s, off, 32);
    const float inv_norm = rsqrtf(s);

    const int half = lane >> 4;     // which K-pair this lane holds
    const int l    = lane & 15;     // row (for A) / column (for B,D)
    const float* xr = x + (size_t)(rowBase + l) * C_DIM;

    v8f c = {};
    for (int k = 0; k < C_DIM; k += 4) {
        v2f a, b;
        a.x = xr[k + 2 * half];
        a.y = xr[k + 2 * half + 1];
        b.x = w1[k + 2 * half];
        b.y = w1[k + 2 * half + 1];
        // (neg_a, A, neg_b, B, c_mod, C, reuse_a, reuse_b)
        c = __builtin_amdgcn_wmma_f32_16x16x4_f32(false, a, false, b,
                                                  (short)0, c, false, false);
    }

    // D layout: VGPR r -> M=r (lanes 0-15) / M=8+r (lanes 16-31); all columns
    // are identical (B broadcast), so column 0 of each half writes its 8 rows.
    if (l == 0) {
        const int r0 = rowBase + half * 8;
        #pragma unroll
        for (int r = 0; r < 8; ++r) {
            float f = tanhf(c[r] * inv_norm);
            fitness[r0 + r] = f;
            unsigned int u = __float_as_uint(f);
            u = (u & 0x80000000u) ? ~u : (u | 0x80000000u); // order-preserving key
            keys[r0 + r] = u;
        }
    }
}

// ---------------------------------------------------------------------------
// Kernel 2: stable rank by counting. blockIdx.y selects a j-chunk; partial
// counts accumulated with one atomicAdd per (i, chunk). The j stream is
// staged into LDS with async global->LDS b128 copies (ASYNCcnt), plus a
// prefetch of the next tile (global_prefetch_b8).
// ---------------------------------------------------------------------------
#define RANK_CHUNKS 8
#define RANK_TILE   2048

__global__ void k_rank(const unsigned int* __restrict__ keys,
                       int* __restrict__ rank) {
    __shared__ unsigned int tile[RANK_TILE];
    const int i  = blockIdx.x * blockDim.x + threadIdx.x;
    const unsigned int ki = keys[i];
    const int j0 = blockIdx.y * (N_PTS / RANK_CHUNKS);
    const int j1 = j0 + (N_PTS / RANK_CHUNKS);

    int r = 0;
    for (int tb = j0; tb < j1; tb += RANK_TILE) {
        __syncthreads();
        // async fill: 256 threads x 2 x 16B = 8KB tile, no VGPR round-trip
        for (int j = threadIdx.x * 4; j < RANK_TILE; j += blockDim.x * 4)
            async_g2l_b128(&keys[tb + j], &tile[j]);
        wait_async0();
        __syncthreads();
        if (tb + RANK_TILE < j1)
            __builtin_prefetch(&keys[tb + RANK_TILE + threadIdx.x], 0, 1);
        for (int j = 0; j < RANK_TILE; ++j) {
            const unsigned int kj = tile[j];
            const int jj = tb + j;
            // stable argsort equivalence: smaller key, or equal key earlier index
            r += (kj < ki) || (kj == ki && jj < i);
        }
    }
    atomicAdd(&rank[i], r);
}

// ---------------------------------------------------------------------------
// Kernel 3: centroid selection: element with rank c*STRIDE is centroid c
// (cent = argsort(fitness)[::STRIDE]).
// ---------------------------------------------------------------------------
__global__ void k_select(const int* __restrict__ rank,
                         const float* __restrict__ fitness,
                         const float* __restrict__ xy,
                         float* __restrict__ cf,
                         float* __restrict__ cxy) {
    const int i = blockIdx.x * blockDim.x + threadIdx.x;
    const int r = rank[i];
    if (r % STRIDE_ == 0) {
        const int slot = r / STRIDE_;   // slot in [0, K_CENT)
        cf[slot]          = fitness[i];
        cxy[2 * slot]     = xy[2 * i];
        cxy[2 * slot + 1] = xy[2 * i + 1];
    }
}

// ---------------------------------------------------------------------------
// Kernel 4: cluster assignment. All K centroids (39 KB) staged into LDS via
// async global->LDS copies; each thread scans them for its point. Also
// accumulates counts and xy sums with fire-and-forget f32 atomics.
// ---------------------------------------------------------------------------
__global__ void k_cluster(const float* __restrict__ xy,
                          const float* __restrict__ fitness,
                          const float* __restrict__ cf,
                          const float* __restrict__ cxy,
                          int*   __restrict__ cluster_i,
                          float* __restrict__ cluster_f,
                          float* __restrict__ counts,
                          float* __restrict__ sum_xy) {
    __shared__ float scx[K_CENT];
    __shared__ float scy[K_CENT];
    __shared__ float scf[K_CENT];
    for (int j = threadIdx.x; j < K_CENT; j += blockDim.x) {
        async_g2l_b32(&cxy[2 * j],     &scx[j]);
        async_g2l_b32(&cxy[2 * j + 1], &scy[j]);
        async_g2l_b32(&cf[j],          &scf[j]);
    }
    wait_async0();
    __syncthreads();

    const int i = blockIdx.x * blockDim.x + threadIdx.x;
    const float px = xy[2 * i];
    const float py = xy[2 * i + 1];
    const float pf = fitness[i];

    float best = INFINITY;
    int bidx = 0;
    for (int j = 0; j < K_CENT; ++j) {
        const float dx = scx[j] - px;
        const float dy = scy[j] - py;
        const float d  = sqrtf(dx * dx + dy * dy) + fabsf(scf[j] - pf);
        if (d < best) { best = d; bidx = j; }   // strict <: argmin keeps first
    }
    cluster_i[i] = bidx;
    cluster_f[i] = (float)bidx;
    atomicAdd(&counts[bidx], 1.0f);
    atomicAdd(&sum_xy[2 * bidx],     px);
    atomicAdd(&sum_xy[2 * bidx + 1], py);
}

// ---------------------------------------------------------------------------
// Kernel 5: feature segment-sum. One point per (block, iteration); lane c
// adds x[p][c] into sum_x[cluster[p]][c]. Coalesced f32 atomics, L2-resident
// 3.4 MB accumulator.
// ---------------------------------------------------------------------------
#define PTS_PER_BLOCK 4
__global__ void k_scatter_x(const float* __restrict__ x,
                            const int* __restrict__ cluster_i,
                            float* __restrict__ sum_x) {
    const int t  = threadIdx.x;       // 0..255 = feature index
    const int p0 = blockIdx.x * PTS_PER_BLOCK;
    #pragma unroll
    for (int q = 0; q < PTS_PER_BLOCK; ++q) {
        const int p = p0 + q;
        const int c = cluster_i[p];
        atomicAdd(&sum_x[(size_t)c * C_DIM + t], x[(size_t)p * C_DIM + t]);
    }
}

// ---------------------------------------------------------------------------
// Kernel 6: finalize: divide sums by max(count, 1).
// ---------------------------------------------------------------------------
__global__ void k_finalize(const float* __restrict__ counts,
                           const float* __restrict__ sum_x,
                           const float* __restrict__ sum_xy,
                           float* __restrict__ out_new_x,
                           float* __restrict__ out_new_xy) {
    const int idx = blockIdx.x * blockDim.x + threadIdx.x;
    if (idx < K_CENT * C_DIM) {
        const int k = idx / C_DIM;
        const float denom = fmaxf(counts[k], 1.0f);
        out_new_x[idx] = sum_x[idx] / denom;
    }
    if (idx < 2 * K_CENT) {
        const int k = idx >> 1;
        const float denom = fmaxf(counts[k], 1.0f);
        out_new_xy[idx] = sum_xy[idx] / denom;
    }
}

// ---------------------------------------------------------------------------
// Host-side launch
// ---------------------------------------------------------------------------
extern "C" void kernel_launch(void* const* d_in, const int* in_sizes, int n_in,
                              void* d_out, int out_size, void* d_ws, size_t ws_size,
                              hipStream_t stream) {
    const float* x  = (const float*)d_in[0];   // N x C
    const float* xy = (const float*)d_in[1];   // N x 2
    const float* w1 = (const float*)d_in[2];   // 1 x C

    // d_out: [new_x (K*C)] [new_xy (2K)] [fitness (N)] [cluster-as-float (N)]
    float* out           = (float*)d_out;
    float* out_new_x     = out;
    float* out_new_xy    = out + (size_t)K_CENT * C_DIM;
    float* out_fitness   = out_new_xy + 2 * K_CENT;
    float* out_cluster   = out_fitness + N_PTS;

    // workspace layout
    unsigned int* keys   = (unsigned int*)d_ws;              // N
    int*   rank          = (int*)(keys + N_PTS);             // N
    float* cf            = (float*)(rank + N_PTS);           // K
    float* cxy           = cf + K_CENT;                      // 2K
    int*   cluster_i     = (int*)(cxy + 2 * K_CENT);         // N
    float* counts        = (float*)(cluster_i + N_PTS);      // K
    float* sum_xy        = counts + K_CENT;                  // 2K
    float* sum_x         = sum_xy + 2 * K_CENT;              // K*C

    // 0) zero rank, then counts+sum_xy+sum_x (contiguous)
    {
        int n1 = N_PTS;
        k_zero<<<(n1 + 255) / 256, 256, 0, stream>>>((unsigned int*)rank, n1);
        int n2 = K_CENT + 2 * K_CENT + K_CENT * C_DIM;
        k_zero<<<(n2 + 255) / 256, 256, 0, stream>>>((unsigned int*)counts, n2);
    }

    // 1) fitness via WMMA: 8 waves/block, 16 rows/wave -> 128 rows/block
    k_fitness<<<N_PTS / 128, 256, 0, stream>>>(x, w1, out_fitness, keys);

    // 2) stable rank (N x N counting, j split into 8 chunks)
    {
        dim3 grid(N_PTS / 256, RANK_CHUNKS);
        k_rank<<<grid, 256, 0, stream>>>(keys, rank);
    }

    // 3) centroid selection
    k_select<<<N_PTS / 256, 256, 0, stream>>>(rank, out_fitness, xy, cf, cxy);

    // 4) cluster assignment + count/xy accumulation
    k_cluster<<<N_PTS / 256, 256, 0, stream>>>(xy, out_fitness, cf, cxy,
                                               cluster_i, out_cluster,
                                               counts, sum_xy);

    // 5) feature segment-sum
    k_scatter_x<<<N_PTS / PTS_PER_BLOCK, C_DIM, 0, stream>>>(x, cluster_i, sum_x);

    // 6) finalize means
    k_finalize<<<(K_CENT * C_DIM + 255) / 256, 256, 0, stream>>>(
        counts, sum_x, sum_xy, out_new_x, out_new_xy);
}